// local_Xinanjiang_89816356094489
// MI455X (gfx1250) — compile-verified
//
#include <hip/hip_runtime.h>
#include <hip/hip_bf16.h>

// ---------------- problem geometry ----------------
#define T_LEN    2048
#define NB       64              // threads per block == batch rows per block (2 waves)
#define NT       32              // time steps staged per LDS chunk
#define NCHUNK   (T_LEN / NT)    // 64
#define ROWDW    99              // 96 data DWORDs + 3 TDM-inserted pad DWORDs (1 per 32)
                                 // lane stride 99 is odd mod 64 banks -> conflict-free
#define REGB     (32 * ROWDW * 4)   // LDS bytes per wave region  (12672)
#define BUFB     (NB * ROWDW * 4)   // LDS bytes per buffer       (25344)
#define CLIPV    100000.0f

typedef unsigned int v4u __attribute__((ext_vector_type(4)));
typedef unsigned int v8u __attribute__((ext_vector_type(8)));

// ---------------- CDNA5 fast paths ----------------

// v_tanh_f32 is a native TRANS op on gfx1250.
__device__ __forceinline__ float tanh_fast(float x) {
#if __has_builtin(__builtin_amdgcn_tanhf)
    return __builtin_amdgcn_tanhf(x);
#elif __has_builtin(__builtin_amdgcn_tanh_f32)
    return __builtin_amdgcn_tanh_f32(x);
#else
    float r;
    asm("v_tanh_f32 %0, %1" : "=v"(r) : "v"(x));
    return r;
#endif
}

__device__ __forceinline__ float heav(float x) {
    return __builtin_fmaf(tanh_fast(10.0f * x), 0.5f, 0.5f);  // (tanh(10x)+1)*0.5
}

__device__ __forceinline__ float clipv(float x) {
    return fminf(fmaxf(x, -CLIPV), CLIPV);
}

// Tensor Data Mover: 2D tile load, groups 2/3 disabled (NULL) => 2D descriptor form.
// Issued once per wave; EXEC is ignored by TDM. Tracked by TENSORcnt.
__device__ __forceinline__ void tdm_load_2d(v4u g0, v8u g1) {
    asm volatile("tensor_load_to_lds %0, %1"
                 :: "s"(g0), "s"(g1)
                 : "memory");
}

__device__ __forceinline__ void wait_tensorcnt0() {
#if __has_builtin(__builtin_amdgcn_s_wait_tensorcnt)
    __builtin_amdgcn_s_wait_tensorcnt(0);
#else
    asm volatile("s_wait_tensorcnt 0x0" ::: "memory");
#endif
}

// Build group-0 descriptor (lds addr + global addr) and fire the TDM.
__device__ __forceinline__ void tdm_issue_chunk(const float* gtile, unsigned lds_byte, v8u g1) {
    const unsigned long long ga = (unsigned long long)(size_t)(const void*)gtile;
    v4u g0;
    g0[0] = 1u;                                            // count=1 (valid), no gather
    g0[1] = lds_byte;                                      // lds_addr
    g0[2] = (unsigned)ga;                                  // global_addr[31:0]
    g0[3] = ((unsigned)(ga >> 32) & 0x01FFFFFFu)           // global_addr[56:32]
          | 0x80000000u;                                   // type = 2 ("image")
    tdm_load_2d(g0, g1);
}

// ---------------- kernel ----------------
__global__ __launch_bounds__(NB, 1)
void xaj_kernel(const float* __restrict__ in,     // [B, T, 3]
                const float* __restrict__ wum_p, const float* __restrict__ wlm_p,
                const float* __restrict__ wdm_p, const float* __restrict__ c_p,
                const float* __restrict__ b_p,   const float* __restrict__ k1_p,
                const float* __restrict__ k2_p,  const float* __restrict__ k3_p,
                float* __restrict__ out)          // [B, T]
{
    __shared__ float lin[2 * NB * ROWDW];         // 50688 B, double-buffered padded tiles

    const int tid  = threadIdx.x;
    const int wave = tid >> 5;                    // wave-uniform
    const int b0   = blockIdx.x * NB;
    const int b    = b0 + tid;

    // ---- fold the 8 scalar params into 3 per-run constants ----
    // reference calls runoff_production(wu, wd, wl, p, wum, wdm, wlm, b, c):
    // wl<->wd and wlm<->wdm are swapped at the call site -- replicated exactly.
    const float wum = wum_p[0], wlm = wlm_p[0], wdm = wdm_p[0];
    const float cc  = c_p[0],   bbv = b_p[0];
    const float k1  = k1_p[0],  k2  = k2_p[0],  k3 = k3_p[0];

    const float wum_ = wum * 19.9f + 0.1f;
    const float wlm_ = wdm * 30.0f + 60.0f;       // receives wdm (swapped)
    const float wdm_ = wlm * 60.0f + 60.0f;       // receives wlm (swapped)
    const float c_   = cc  * 0.19f + 0.01f;
    const float b_   = bbv * 0.30f + 0.10f;
    const float wt   = wum_ + wlm_ + wdm_;
    const float inv  = 1.0f / wt;
    const float sA   = c_ * inv * inv;            // s = sA*wu^2 + sB*wd^2
    const float sB   = b_ * inv * inv;

    const float k1_  = k1 * 0.69f + 0.01f;
    const float k2_  = k2 * 0.69f + 0.01f;
    const float k3_  = k3 * 0.89f + 0.01f;
    const float qc   = k1_ + 0.5f * k2_ * (1.0f - k1_)
                     + 0.25f * k3_ * (1.0f - k1_) * (1.0f - k2_);   // total_q = qc * runoff

    // ---- constant part of the TDM descriptor (group 1) ----
    // 2D tile: dim0 = 96 contiguous floats (32 steps x 3), dim1 = 32 batch rows.
    // data_size=4B, pad 1 DWORD after every 32 data DWORDs (rows: 96 -> 99 DW stride).
    v8u g1;
    g1[0] = (2u << 16)            // data_size = 4 bytes
          | (1u << 20)            // pad_enable
          | (4u << 22);           // pad_interval: 32 DWORDs   (pad_amount=0 -> 1 DWORD)
                                  // workgroup_mask = 0 (not in a cluster)
    g1[1] = (96u << 16);          // tensor_dim0 = 96  (bits 79:48, low half)
    g1[2] = (32u << 16);          // tensor_dim0 hi = 0 ; tensor_dim1 = 32 (bits 111:80)
    g1[3] = (96u << 16);          // tensor_dim1 hi = 0 ; tile_dim0 = 96
    g1[4] = 32u;                  // tile_dim1 = 32 ; tile_dim2 = 0 (2D)
    g1[5] = (unsigned)(T_LEN * 3);// tensor_dim0_stride = 6144 elements (row-to-row)
    g1[6] = 0u;                   // stride hi / dim1_stride lo (unused for 2D)
    g1[7] = 0u;

    const unsigned lds_base = (unsigned)(size_t)(void*)lin;   // low 32b flat == LDS offset
    const unsigned lds_wave = lds_base + (unsigned)wave * REGB;
    const float*   gwave    = in + (size_t)(b0 + wave * 32) * (T_LEN * 3);

    // ---- prologue: stage chunk 0 (each wave DMAs its own 32 rows) ----
    tdm_issue_chunk(gwave, lds_wave + 0u * BUFB, g1);
    wait_tensorcnt0();

    float wu = 0.0f, wl = 0.0f, wd = 0.0f;
    float q0 = 0.0f, q1 = 0.0f, q2 = 0.0f;

    for (int ch = 0; ch < NCHUNK; ++ch) {
        // ---- fire DMA for chunk ch+1 into the other buffer; overlaps with compute ----
        if (ch + 1 < NCHUNK) {
            tdm_issue_chunk(gwave + (size_t)(ch + 1) * (NT * 3),
                            lds_wave + (unsigned)(((ch + 1) & 1) * BUFB), g1);
        }

        const float* lrow = lin + (ch & 1) * (NB * ROWDW) + tid * ROWDW;
        float*       orow = out + (size_t)b * T_LEN + ch * NT;

        #pragma unroll
        for (int tt = 0; tt < NT; ++tt) {
            const int i0 = tt * 3;            // padded LDS index: idx + idx/32
            const int i2 = tt * 3 + 2;        // (compile-time constants when unrolled)
            const float pet = lrow[i0 + (i0 >> 5)];
            const float p   = lrow[i2 + (i2 >> 5)];

            // evapotranspiration; heaviside(-x) == 1 - heaviside(x) (tanh is odd)
            const float h1   = heav(wu - pet);
            const float et1  = wu + h1 * (pet - wu);
            const float d1   = pet - et1;
            const float rem  = heav(d1) * d1;
            const float h3   = heav(rem - wl);
            const float et22 = rem + h3 * (wl - rem);
            const float et2  = heav(rem) * et22;
            const float r2   = rem - et2;
            const float h5   = heav(r2 - wd);
            const float et33 = r2 + h5 * (wd - r2);
            const float et3  = heav(r2) * et33;

            wu += clipv(p - et1);
            wl += clipv(p - et1 - et2);
            wd += clipv(p - et1 - et2 - et3);

            // fused runoff + partition + routing (off the recurrence chain)
            const float s  = sA * wu * wu + sB * wd * wd;
            const float ps = p - s;
            const float tq = qc * (heav(ps) * ps);

            if      ((tt & 3) == 0) q0 = tq;
            else if ((tt & 3) == 1) q1 = tq;
            else if ((tt & 3) == 2) q2 = tq;
            else {
                *(float4*)(orow + (tt - 3)) = make_float4(q0, q1, q2, tq);  // b128 store
            }
        }

        // wave-private tiles: only our own TENSORcnt matters -- no __syncthreads needed
        wait_tensorcnt0();
    }
}

// ---------------- launch ----------------
extern "C" void kernel_launch(void* const* d_in, const int* in_sizes, int n_in,
                              void* d_out, int out_size, void* d_ws, size_t ws_size,
                              hipStream_t stream) {
    (void)n_in; (void)out_size; (void)d_ws; (void)ws_size;
    const float* in  = (const float*)d_in[0];
    const float* wum = (const float*)d_in[1];
    const float* wlm = (const float*)d_in[2];
    const float* wdm = (const float*)d_in[3];
    const float* c   = (const float*)d_in[4];
    const float* b   = (const float*)d_in[5];
    const float* k1  = (const float*)d_in[6];
    const float* k2  = (const float*)d_in[7];
    const float* k3  = (const float*)d_in[8];
    float* out = (float*)d_out;

    const int B = in_sizes[0] / (T_LEN * 3);   // 4096
    const int grid = B / NB;                   // 64 blocks x 2 waves

    hipLaunchKernelGGL(xaj_kernel, dim3(grid), dim3(NB), 0, stream,
                       in, wum, wlm, wdm, c, b, k1, k2, k3, out);
}